// FeatureExtractor_79448305041538
// MI455X (gfx1250) — compile-verified
//
#include <hip/hip_runtime.h>
#include <math.h>

// Problem constants (from the reference):
//   B=8, C=64, L=8192, H=32, UNIT=512, W=544, nblocks=16
#define B_SZ   8
#define C_CH   64
#define L_LEN  8192
#define H_BAND 32
#define UNIT_L 512
#define NBLK   16
#define NSTRIP 32          // 512 / 16 row strips per block
#define NPOS   48          // positions staged per strip: rows [w0,w0+16) + cols [w0,w0+48)
#define LSTRIDE 68         // LDS row stride in dwords; 68 % 64 == 4 -> conflict-free frag loads
#define EPS_F  1e-12f

typedef float v2f __attribute__((ext_vector_type(2)));
typedef float v8f __attribute__((ext_vector_type(8)));

__global__ __launch_bounds__(96)
void FeatureExtractor_79448305041538_kernel(const float* __restrict__ feat,
                                            float* __restrict__ out) {
    __shared__ float Xs[NPOS * LSTRIDE];   // centered window, [pos][chan], padded stride
    __shared__ float normS[NPOS];          // per-position channel-vector norms

    const int blk   = blockIdx.x;          // 0 .. 4095
    const int strip = blk & (NSTRIP - 1);  // row strip within block
    const int n     = (blk >> 5) & (NBLK - 1);
    const int b     = blk >> 9;
    const int start = n * UNIT_L;
    const int w0    = strip * 16;          // first row (window-local) of this strip

    const int tid = threadIdx.x;

    // ---- Stage 64x48 f32 sub-window into LDS (zero-pad past L) ----
    const float* fb = feat + (size_t)b * C_CH * L_LEN;
    for (int idx = tid; idx < C_CH * NPOS; idx += 96) {
        int c = idx / NPOS;
        int p = idx - c * NPOS;
        int g = start + w0 + p;            // global position in sequence
        float v = 0.0f;
        if (g < L_LEN) v = fb[(size_t)c * L_LEN + g];
        Xs[p * LSTRIDE + c] = v;
    }
    __syncthreads();

    // ---- Channel-mean subtraction + norms: one thread per position ----
    if (tid < NPOS) {
        float s = 0.0f;
        for (int c = 0; c < C_CH; ++c) s += Xs[tid * LSTRIDE + c];
        float mu = s * (1.0f / 64.0f);
        float ss = 0.0f;
        for (int c = 0; c < C_CH; ++c) {
            float x = Xs[tid * LSTRIDE + c] - mu;
            Xs[tid * LSTRIDE + c] = x;
            ss += x * x;
        }
        normS[tid] = sqrtf(ss);
    }
    __syncthreads();

    // ---- One 16x16 Gram tile per wave: G = Xc^T * Xc over K=64 (16x K=4 WMMA) ----
    const int wave  = tid >> 5;            // column tile j = 0,1,2
    const int lane  = tid & 31;
    const int nn    = lane & 15;           // N (B/C/D column), also A row M
    const int khalf = lane >> 4;           // 0: K={0,1}; 1: K={2,3} (32-bit A/B layout)

    const int aRow = nn;                   // A fragment: row M = lane%16
    const int bRow = wave * 16 + nn;       // B fragment: column position

    v8f acc = {};
#pragma unroll
    for (int k0 = 0; k0 < 16; ++k0) {
        int c = (k0 << 2) + (khalf << 1);
        v2f av = *(const v2f*)&Xs[aRow * LSTRIDE + c];   // ds_load_b64, bank-conflict free
        v2f bv = *(const v2f*)&Xs[bRow * LSTRIDE + c];
        acc = __builtin_amdgcn_wmma_f32_16x16x4_f32(
            /*neg_a=*/false, av, /*neg_b=*/false, bv,
            /*c_mod=*/(short)0, acc, /*reuse_a=*/false, /*reuse_b=*/false);
    }

    // ---- Normalize, band-extract, scatter: d = v - t - 1 in [0,32) ----
    const float ncol = normS[bRow];
    const int   vloc = wave * 16 + nn;     // window-local column index
    const int   gcol = start + w0 + vloc;  // global column (mask if >= L)
#pragma unroll
    for (int r = 0; r < 8; ++r) {
        int m = r + (khalf << 3);          // C/D layout: M = r + 8*(lane/16)
        int d = vloc - m - 1;              // band offset
        if ((unsigned)d < (unsigned)H_BAND) {
            float denom = fmaxf(normS[m] * ncol, EPS_F);
            float val   = acc[r] / denom;
            if (gcol >= L_LEN) val = 0.0f; // zero past end of sequence
            int t = w0 + m;                // row within block -> output l = start + t
            out[((size_t)b * H_BAND + d) * L_LEN + (start + t)] = val;
        }
    }
}

extern "C" void kernel_launch(void* const* d_in, const int* in_sizes, int n_in,
                              void* d_out, int out_size, void* d_ws, size_t ws_size,
                              hipStream_t stream) {
    (void)in_sizes; (void)n_in; (void)out_size; (void)d_ws; (void)ws_size;
    const float* feat = (const float*)d_in[0];
    float* out = (float*)d_out;
    // 8 batches * 16 blocks * 32 strips = 4096 workgroups, 3 waves each
    dim3 grid(B_SZ * NBLK * NSTRIP), block(96);
    FeatureExtractor_79448305041538_kernel<<<grid, block, 0, stream>>>(feat, out);
}